// SS2D_66314295050840
// MI455X (gfx1250) — compile-verified
//
#include <hip/hip_runtime.h>
#include <hip/hip_bf16.h>
#include <math.h>

// Problem constants (from reference)
#define BATCH   4
#define KGRP    4
#define DIN     384      // D_INNER
#define NST     16       // D_STATE
#define RNK     12       // DT_RANK
#define HH      32
#define WW      32
#define LL      1024     // H*W
#define CROWS   44       // DT_RANK + 2*D_STATE
#define CPAD    48       // padded to 3 x 16 WMMA tiles
#define NCH     (BATCH * KGRP * DIN)   // 6144 scan channels
#define NCHUNK  8
#define CHUNK   128      // LL / NCHUNK

typedef float v2f __attribute__((ext_vector_type(2)));
typedef float v8f __attribute__((ext_vector_type(8)));

// Map scan position l of direction k to row-major index in the HxW plane.
__device__ __forceinline__ int scan_index(int k, int l) {
    int lr = (k & 2) ? (LL - 1 - l) : l;
    if (k & 1) return (lr & (WW - 1)) * WW + (lr >> 5);
    return lr;
}

// ---------------------------------------------------------------------------
// Kernel 1: x_dbl[b,k,:,l] = Wproj[k] @ xs[b,k,:,l]  via V_WMMA_F32_16X16X4_F32
// Output stored transposed/padded: xdbl[(b*K+k), l, 0..47].
// A-row index is CLAMPED (not predicated) for rows 44..47: their D rows are
// garbage in padded slots that are never read, and clamping keeps all loads
// unconditional (no EXEC toggling around the WMMA pipeline).
// ---------------------------------------------------------------------------
__global__ void __launch_bounds__(32) ss2d_proj_wmma(
    const float* __restrict__ x,      // (B, DIN, H, W)
    const float* __restrict__ wproj,  // (K, 44, 384)
    float* __restrict__ xdbl)         // (B*K, L, 48)
{
    const int lane = threadIdx.x;     // full wave active (EXEC all 1s)
    const int half = lane >> 4;
    const int lr   = lane & 15;

    int t = blockIdx.x;
    const int nt = t & 63;  t >>= 6;   // 64 N-tiles (L/16)
    const int mt = t % 3;   t /= 3;    // 3  M-tiles (48/16)
    const int k  = t & 3;   t >>= 2;
    const int b  = t;

    const int col  = nt * 16 + lr;          // l position for B-operand
    const int row  = mt * 16 + lr;          // c row for A-operand
    const int rowc = (row < CROWS) ? row : (CROWS - 1);   // clamp, see above
    const int midx = scan_index(k, col);    // cross-scan gather index

    const float* wrow = wproj + (k * CROWS + rowc) * DIN;
    const float* xcol = x + (size_t)b * DIN * LL + midx;

    v8f acc0 = {0.f,0.f,0.f,0.f,0.f,0.f,0.f,0.f};
    v8f acc1 = {0.f,0.f,0.f,0.f,0.f,0.f,0.f,0.f};

    for (int kk = 0; kk < DIN; kk += 8) {
        {
            const int j = kk + 2 * half;    // A/B: lanes<16 K={0,1}, >=16 K={2,3}
            v2f a, bm;
            a.x  = wrow[j];
            a.y  = wrow[j + 1];
            bm.x = xcol[(size_t)(j)     * LL];
            bm.y = xcol[(size_t)(j + 1) * LL];
            acc0 = __builtin_amdgcn_wmma_f32_16x16x4_f32(
                false, a, false, bm, (short)0, acc0, false, false);
        }
        {
            const int j = kk + 4 + 2 * half;
            v2f a, bm;
            a.x  = wrow[j];
            a.y  = wrow[j + 1];
            bm.x = xcol[(size_t)(j)     * LL];
            bm.y = xcol[(size_t)(j + 1) * LL];
            acc1 = __builtin_amdgcn_wmma_f32_16x16x4_f32(
                false, a, false, bm, (short)0, acc1, false, false);
        }
    }
    v8f acc = acc0 + acc1;

    // D layout: VGPR v -> M = v + 8*half, N = lr. 8 contiguous floats/lane.
    float* drow = xdbl + ((size_t)(b * KGRP + k) * LL + col) * CPAD
                       + mt * 16 + half * 8;
    #pragma unroll
    for (int v = 0; v < 8; ++v) drow[v] = acc[v];
}

// ---------------------------------------------------------------------------
// Chunked selective scan (linear recurrence h' = dA*h + du*B):
//   Pass A: per (chunk, channel) scan from h=0 -> local state hloc + decay P
//   Pass B: per (channel, state) serial 8-step combine -> h_in per chunk
//   Pass C: per (chunk, channel) rescan from h_in, emit y
// Cuts the dependent chain 1024 -> 128+8+128 and gives 8x scan occupancy.
// ---------------------------------------------------------------------------

// channel decode helper: ch = (b*K + k)*DIN + d
__device__ __forceinline__ void ch_decode(int ch, int& b, int& k, int& d, int& kd, int& bk) {
    b  = ch / (KGRP * DIN);
    const int r = ch - b * (KGRP * DIN);
    k  = r / DIN;
    d  = r - k * DIN;
    kd = k * DIN + d;
    bk = __builtin_amdgcn_readfirstlane(b * KGRP + k);  // waves never straddle (b,k)
}

__device__ __forceinline__ float softplus_f(float v) {
    return (v > 20.f) ? v : __logf(1.f + __expf(v));
}

__global__ void __launch_bounds__(256) ss2d_scan_partial(
    const float* __restrict__ x,      // (B, DIN, L)
    const float* __restrict__ xdbl,   // (B*K, L, 48)
    const float* __restrict__ dtw,    // (K, DIN, RNK)
    const float* __restrict__ dtb,    // (K, DIN)
    const float* __restrict__ Alogs,  // (K*DIN, NST)
    float* __restrict__ Pout,         // (NCH, NCHUNK, NST)
    float* __restrict__ Hout)         // (NCH, NCHUNK, NST)
{
    const int g  = blockIdx.x * 256 + threadIdx.x;
    const int ch = g % NCH;
    const int c  = g / NCH;           // chunk id, uniform per block (NCH%256==0)
    int b, k, d, kd, bk;
    ch_decode(ch, b, k, d, kd, bk);

    float A_[NST], P[NST], h[NST];
    #pragma unroll
    for (int n = 0; n < NST; ++n) { A_[n] = -__expf(Alogs[kd * NST + n]); P[n] = 1.f; h[n] = 0.f; }
    float wdt[RNK];
    #pragma unroll
    for (int rr = 0; rr < RNK; ++rr) wdt[rr] = dtw[kd * RNK + rr];
    const float bias = dtb[kd];

    const float* xch  = x    + (size_t)(b * DIN + d) * LL;
    const float* xrow = xdbl + (size_t)bk * LL * CPAD;
    const int l0 = c * CHUNK;

    for (int l = l0; l < l0 + CHUNK; ++l) {
        const float* row = xrow + l * CPAD;     // uniform 48-float row
        float dtr = bias;
        #pragma unroll
        for (int rr = 0; rr < RNK; ++rr) dtr += row[rr] * wdt[rr];
        const float dt = softplus_f(dtr);
        const float du = dt * xch[scan_index(k, l)];
        #pragma unroll
        for (int n = 0; n < NST; ++n) {
            const float dA = __expf(dt * A_[n]);
            h[n] = h[n] * dA + du * row[RNK + n];
            P[n] *= dA;
        }
    }
    float* pp = Pout + (size_t)(ch * NCHUNK + c) * NST;
    float* hp = Hout + (size_t)(ch * NCHUNK + c) * NST;
    #pragma unroll
    for (int n = 0; n < NST; ++n) { pp[n] = P[n]; hp[n] = h[n]; }
}

__global__ void __launch_bounds__(256) ss2d_scan_combine(
    const float* __restrict__ Pin,    // (NCH, NCHUNK, NST)
    const float* __restrict__ Hloc,   // (NCH, NCHUNK, NST)
    float* __restrict__ Hin)          // (NCH, NCHUNK, NST)
{
    const int g  = blockIdx.x * 256 + threadIdx.x;  // (channel, state)
    const int ch = g / NST;
    const int n  = g - ch * NST;
    float hc = 0.f;
    #pragma unroll
    for (int c = 0; c < NCHUNK; ++c) {
        const size_t idx = (size_t)(ch * NCHUNK + c) * NST + n;
        Hin[idx] = hc;
        hc = hc * Pin[idx] + Hloc[idx];
    }
}

__global__ void __launch_bounds__(256) ss2d_scan_final(
    const float* __restrict__ x,      // (B, DIN, L)
    const float* __restrict__ xdbl,   // (B*K, L, 48)
    const float* __restrict__ dtw,    // (K, DIN, RNK)
    const float* __restrict__ dtb,    // (K, DIN)
    const float* __restrict__ Alogs,  // (K*DIN, NST)
    const float* __restrict__ Dsv,    // (K*DIN)
    const float* __restrict__ Hin,    // (NCH, NCHUNK, NST)
    float* __restrict__ ys)           // (B*K, L, DIN)
{
    const int g  = blockIdx.x * 256 + threadIdx.x;
    const int ch = g % NCH;
    const int c  = g / NCH;
    int b, k, d, kd, bk;
    ch_decode(ch, b, k, d, kd, bk);

    float A_[NST], h[NST];
    const float* hip_ = Hin + (size_t)(ch * NCHUNK + c) * NST;
    #pragma unroll
    for (int n = 0; n < NST; ++n) { A_[n] = -__expf(Alogs[kd * NST + n]); h[n] = hip_[n]; }
    float wdt[RNK];
    #pragma unroll
    for (int rr = 0; rr < RNK; ++rr) wdt[rr] = dtw[kd * RNK + rr];
    const float bias = dtb[kd];
    const float Dval = Dsv[kd];

    const float* xch  = x    + (size_t)(b * DIN + d) * LL;
    const float* xrow = xdbl + (size_t)bk * LL * CPAD;
    float*       ycol = ys   + (size_t)bk * LL * DIN + d;
    const int l0 = c * CHUNK;

    for (int l = l0; l < l0 + CHUNK; ++l) {
        const float* row = xrow + l * CPAD;
        float dtr = bias;
        #pragma unroll
        for (int rr = 0; rr < RNK; ++rr) dtr += row[rr] * wdt[rr];
        const float dt = softplus_f(dtr);
        const float u  = xch[scan_index(k, l)];
        const float du = dt * u;
        float acc = 0.f;
        #pragma unroll
        for (int n = 0; n < NST; ++n) {
            const float dA = __expf(dt * A_[n]);
            h[n] = h[n] * dA + du * row[RNK + n];
            acc += h[n] * row[RNK + NST + n];
        }
        ycol[(size_t)l * DIN] = acc + Dval * u;   // coalesced across the wave
    }
}

// ---------------------------------------------------------------------------
// Kernel 5: cross-merge + LayerNorm over D + exact GELU.
// One wave per (b,l) site; lane handles 12 d-values; wave32 shfl reduction.
// ---------------------------------------------------------------------------
__global__ void __launch_bounds__(256) ss2d_merge_ln_gelu(
    const float* __restrict__ ys,     // (B*K, L, DIN)
    const float* __restrict__ lnw,
    const float* __restrict__ lnb,
    float* __restrict__ out)          // (B, DIN, H, W)
{
    const int lane = threadIdx.x & 31;
    const int wv   = threadIdx.x >> 5;
    const int site = blockIdx.x * 8 + wv;   // b*L + l
    const int b    = site >> 10;
    const int l    = site & (LL - 1);
    const int m    = (l & 31) * 32 + (l >> 5);  // transpose mapping for k=1/3

    const size_t s0 = ((size_t)(b * 4 + 0) * LL + l)            * DIN;
    const size_t s1 = ((size_t)(b * 4 + 1) * LL + m)            * DIN;
    const size_t s2 = ((size_t)(b * 4 + 2) * LL + (LL - 1 - l)) * DIN;
    const size_t s3 = ((size_t)(b * 4 + 3) * LL + (LL - 1 - m)) * DIN;

    float v[12];
    float s = 0.f, s2a = 0.f;
    #pragma unroll
    for (int j = 0; j < 12; ++j) {
        const int d = lane + 32 * j;
        const float val = ys[s0 + d] + ys[s1 + d] + ys[s2 + d] + ys[s3 + d];
        v[j] = val; s += val; s2a += val * val;
    }
    #pragma unroll
    for (int off = 16; off > 0; off >>= 1) {
        s   += __shfl_xor(s,   off, 32);
        s2a += __shfl_xor(s2a, off, 32);
    }
    const float mu  = s   * (1.f / DIN);
    const float var = s2a * (1.f / DIN) - mu * mu;
    const float rs  = rsqrtf(var + 1e-5f);
    #pragma unroll
    for (int j = 0; j < 12; ++j) {
        const int d = lane + 32 * j;
        const float tt = (v[j] - mu) * rs * lnw[d] + lnb[d];
        const float ge = 0.5f * tt * (1.f + erff(tt * 0.70710678118654752f));
        out[(size_t)(b * DIN + d) * LL + l] = ge;
    }
}

// ---------------------------------------------------------------------------
extern "C" void kernel_launch(void* const* d_in, const int* in_sizes, int n_in,
                              void* d_out, int out_size, void* d_ws, size_t ws_size,
                              hipStream_t stream) {
    (void)in_sizes; (void)n_in; (void)out_size; (void)ws_size;
    const float* x     = (const float*)d_in[0];  // (4,384,32,32)
    const float* wproj = (const float*)d_in[1];  // (4,44,384)
    const float* dtw   = (const float*)d_in[2];  // (4,384,12)
    const float* dtb   = (const float*)d_in[3];  // (4,384)
    const float* Alog  = (const float*)d_in[4];  // (1536,16)
    const float* Dsv   = (const float*)d_in[5];  // (1536)
    const float* lnw   = (const float*)d_in[6];  // (384)
    const float* lnb   = (const float*)d_in[7];  // (384)
    float* out = (float*)d_out;

    // Workspace layout (floats):
    float* xdbl = (float*)d_ws;                               //  786432 (3 MB)
    float* ys   = xdbl + (size_t)BATCH * KGRP * LL * CPAD;    // 6291456 (24 MB)
    float* Pbuf = ys   + (size_t)BATCH * KGRP * LL * DIN;     //  786432 (3 MB)
    float* Hloc = Pbuf + (size_t)NCH * NCHUNK * NST;          //  786432 (3 MB)
    float* Hin  = Hloc + (size_t)NCH * NCHUNK * NST;          //  786432 (3 MB)

    ss2d_proj_wmma<<<BATCH * KGRP * 3 * 64, 32, 0, stream>>>(x, wproj, xdbl);
    ss2d_scan_partial<<<(NCH * NCHUNK) / 256, 256, 0, stream>>>(
        x, xdbl, dtw, dtb, Alog, Pbuf, Hloc);
    ss2d_scan_combine<<<(NCH * NST) / 256, 256, 0, stream>>>(Pbuf, Hloc, Hin);
    ss2d_scan_final<<<(NCH * NCHUNK) / 256, 256, 0, stream>>>(
        x, xdbl, dtw, dtb, Alog, Dsv, Hin, ys);
    ss2d_merge_ln_gelu<<<(BATCH * LL) / 8, 256, 0, stream>>>(ys, lnw, lnb, out);
}